// PointNet2Encoder_1022202217079
// MI455X (gfx1250) — compile-verified
//
#include <hip/hip_runtime.h>
#include <hip/hip_bf16.h>
#include <stdint.h>

// ---------------------------------------------------------------------------
// PointNet++ encoder for MI455X (gfx1250).
// fp32 WMMA (v_wmma_f32_16x16x4_f32) for every MLP GEMM; fused MLP chains with
// LDS ping-pong activations (CDNA5: 320KB LDS/WGP); max-pool via monotonic
// int atomicMax on non-negative (post-ReLU) floats.
// Inner GEMM loop is branchless (clamped-column weight loads) with running
// pointers so WMMAs are not serialized behind exec-masked loads.
// ---------------------------------------------------------------------------

typedef __attribute__((ext_vector_type(2))) float v2f;
typedef __attribute__((ext_vector_type(8))) float v8f;

struct MlpDesc {
  int nlayers;
  int cin[4];
  int cout[4];
  const float* w[4];
  const float* b[4];
};

__device__ __forceinline__ int dceil4(int x) { return (x + 3) & ~3; }

// ----------------------------- utility kernels -----------------------------

__global__ void zero_kernel(float* __restrict__ p, int n) {
  int i = blockIdx.x * blockDim.x + threadIdx.x;
  if (i < n) p[i] = 0.f;
}

// ------------------------------- FPS ---------------------------------------
// One block per batch. dist[] lives in dynamic LDS. Tie-break = first index
// (matches jnp.argmax). Outputs the "far" index *before* each update step.
__global__ void fps_kernel(const float* __restrict__ xyz, int N, int npoint,
                           int* __restrict__ outIdx) {
  extern __shared__ float dist[];
  __shared__ float rmax[256];
  __shared__ int rarg[256];
  const int b = blockIdx.x;
  const float* X = xyz + (size_t)b * N * 3;
  const int tid = threadIdx.x, nthr = blockDim.x;

  for (int i = tid; i < N; i += nthr) dist[i] = 1e10f;
  __syncthreads();

  int far = 0;
  for (int j = 0; j < npoint; ++j) {
    if (tid == 0) outIdx[b * npoint + j] = far;
    const float cx = X[far * 3 + 0];
    const float cy = X[far * 3 + 1];
    const float cz = X[far * 3 + 2];
    float bv = -1.f;
    int bi = 0;
    for (int i = tid; i < N; i += nthr) {
      const float dx = X[i * 3 + 0] - cx;
      const float dy = X[i * 3 + 1] - cy;
      const float dz = X[i * 3 + 2] - cz;
      const float d = dx * dx + dy * dy + dz * dz;
      const float nd = fminf(dist[i], d);
      dist[i] = nd;
      if (nd > bv) { bv = nd; bi = i; }   // strided ascending -> first-max kept
    }
    rmax[tid] = bv; rarg[tid] = bi;
    __syncthreads();
    for (int s = nthr >> 1; s > 0; s >>= 1) {
      if (tid < s) {
        const float ov = rmax[tid + s];
        const int oi = rarg[tid + s];
        if (ov > rmax[tid] || (ov == rmax[tid] && oi < rarg[tid])) {
          rmax[tid] = ov; rarg[tid] = oi;
        }
      }
      __syncthreads();
    }
    far = rarg[0];
    __syncthreads();
  }
}

// ----------------------------- gather centers ------------------------------
__global__ void gather_xyz_kernel(const float* __restrict__ pc,
                                  const int* __restrict__ fidx, int N, int S,
                                  int BS, float* __restrict__ out) {
  int g = blockIdx.x * blockDim.x + threadIdx.x;
  if (g >= BS) return;
  const int b = g / S;
  const int p = fidx[g];
  const float* src = pc + ((size_t)b * N + p) * 3;
  out[g * 3 + 0] = src[0];
  out[g * 3 + 1] = src[1];
  out[g * 3 + 2] = src[2];
}

// ------------------------------ ball query ---------------------------------
// Matches: stable argsort of (mask?0:1) -> first `nsample` in-radius indices
// in ascending index order; empty slots filled with slot0 (or 0 if no hits).
__global__ void ball_query_kernel(const float* __restrict__ pc,
                                  const float* __restrict__ centers, int N,
                                  int S, int BS, int nsample, float r2,
                                  int* __restrict__ idx) {
  int g = blockIdx.x * blockDim.x + threadIdx.x;
  if (g >= BS) return;
  const int b = g / S;
  const float* X = pc + (size_t)b * N * 3;
  const float qx = centers[g * 3 + 0];
  const float qy = centers[g * 3 + 1];
  const float qz = centers[g * 3 + 2];
  int cnt = 0;
  const long long base = (long long)g * nsample;
  for (int n = 0; n < N && cnt < nsample; ++n) {
    const float dx = X[n * 3 + 0] - qx;
    const float dy = X[n * 3 + 1] - qy;
    const float dz = X[n * 3 + 2] - qz;
    if (dx * dx + dy * dy + dz * dz < r2) idx[base + (cnt++)] = n;
  }
  const int fill = (cnt > 0) ? idx[base] : 0;
  for (; cnt < nsample; ++cnt) idx[base + cnt] = fill;
}

// ------------------------------ grouping -----------------------------------
// Row (b,s,k): [ pc[gidx] - center | pts[gidx] ] -> (3+Cp) features.
__global__ void group_feats_kernel(const float* __restrict__ pc,
                                   const float* __restrict__ pts,
                                   const float* __restrict__ centers,
                                   const int* __restrict__ gidx, int B, int N,
                                   int S, int ns, int Cp,
                                   float* __restrict__ out) {
  long long row = (long long)blockIdx.x * blockDim.x + threadIdx.x;
  const long long total = (long long)B * S * ns;
  if (row >= total) return;
  const int g = (int)(row / ns);
  const int b = g / S;
  const int n = gidx[row];
  const int C = 3 + Cp;
  float* o = out + row * C;
  const float* p = pc + ((size_t)b * N + n) * 3;
  const float* c = centers + (size_t)g * 3;
  o[0] = p[0] - c[0];
  o[1] = p[1] - c[1];
  o[2] = p[2] - c[2];
  if (Cp > 0) {
    const float* f = pts + ((size_t)b * N + n) * Cp;
    for (int cc = 0; cc < Cp; ++cc) o[3 + cc] = f[cc];
  }
}

// SA5 input rows: (b, i<576): [cx(b,i) | cp(b,i)] with cx = concat(nx2, nx1),
// cp = concat(p1b, p2). 259 features per row.
__global__ void build_feats5_kernel(const float* __restrict__ nx2,
                                    const float* __restrict__ p1b,
                                    const float* __restrict__ nx1,
                                    const float* __restrict__ p2,
                                    float* __restrict__ out) {
  int row = blockIdx.x * blockDim.x + threadIdx.x;  // 8*576
  if (row >= 8 * 576) return;
  const int b = row / 576;
  const int i = row % 576;
  const float* xs;
  const float* ps;
  if (i < 64) {
    xs = nx2 + (size_t)(b * 64 + i) * 3;
    ps = p1b + (size_t)(b * 64 + i) * 256;
  } else {
    const int k = i - 64;
    xs = nx1 + (size_t)(b * 512 + k) * 3;
    ps = p2 + (size_t)(b * 512 + k) * 256;
  }
  float* o = out + (size_t)row * 259;
  o[0] = xs[0]; o[1] = xs[1]; o[2] = xs[2];
  for (int c = 0; c < 256; ++c) o[3 + c] = ps[c];
}

// feat = concat(p3 (8,512), p4 (8,1024)) -> (8,1536)
__global__ void build_feat_kernel(const float* __restrict__ p3,
                                  const float* __restrict__ p4,
                                  float* __restrict__ feat) {
  int i = blockIdx.x * blockDim.x + threadIdx.x;
  if (i >= 8 * 1536) return;
  const int b = i / 1536, c = i % 1536;
  feat[i] = (c < 512) ? p3[b * 512 + c] : p4[b * 1024 + (c - 512)];
}

// d_out = [ 24 zeros (x4) | fc result (8,256) ]
__global__ void finalize_kernel(const float* __restrict__ fco,
                                float* __restrict__ out) {
  int i = blockIdx.x * blockDim.x + threadIdx.x;
  if (i < 24) out[i] = 0.f;
  if (i < 2048) out[24 + i] = fco[i];
}

// --------------------------- fused MLP + maxpool ----------------------------
// Processes `tile` rows per block. Activations live in two LDS buffers of
// tile*cmax floats. Each 32-lane wave computes 16x16 output tiles with
// v_wmma_f32_16x16x4_f32, looping K in steps of 4.
//   * A-side K padding is zero-filled in LDS.
//   * B-fragment loads are branchless: out-of-range columns load a clamped
//     in-bounds column; those lanes' results are never stored.
//   * Main K loop uses running pointers (no 64-bit muls, no exec masking);
//     the final (possibly padded) K step is peeled with clamped row indices.
// Final layer max-pools groups of `pool` consecutive rows into `out` via
// int atomicMax (all values >= 0 after ReLU; out pre-zeroed).
__global__ void fused_mlp_pool(const float* __restrict__ in, int R, int tile,
                               MlpDesc md, float* __restrict__ out, int pool,
                               int cmax) {
  extern __shared__ float smem[];
  float* cur = smem;
  float* nxt = smem + (size_t)tile * cmax;
  const int tid = threadIdx.x;
  const int nthr = blockDim.x;
  const int lane = tid & 31;
  const int wave = tid >> 5;
  const int nwaves = nthr >> 5;
  const int row0 = blockIdx.x * tile;
  const int khalf = (lane >> 4) << 1;  // 0 for lanes 0-15, 2 for 16-31
  const int l15 = lane & 15;
  const int rhalf = (lane >> 4) << 3;  // D-fragment row offset

  // Load input tile into LDS, zero-pad K to multiple of 4 and rows >= R.
  {
    const int C0 = md.cin[0];
    const int K4 = dceil4(C0);
    for (int i = tid; i < tile * K4; i += nthr) {
      const int r = i / K4, c = i % K4;
      const int gr = row0 + r;
      float v = 0.f;
      if (gr < R && c < C0) v = in[(size_t)gr * C0 + c];
      cur[i] = v;
    }
  }

  for (int l = 0; l < md.nlayers; ++l) {
    const int Cin = md.cin[l];
    const int Kp = dceil4(Cin);
    const int Cout = md.cout[l];
    const int Ko = dceil4(Cout);
    const bool last = (l == md.nlayers - 1);
    const float* __restrict__ W = md.w[l];
    const float* __restrict__ Bs = md.b[l];

    // Warm WGP$/L2 with this layer's weights (bounded; global_prefetch_b8).
    {
      const size_t nW = (size_t)Cin * Cout;
      const size_t lim = nW < 16384 ? nW : 16384;
      for (size_t i = (size_t)tid * 16; i < lim; i += (size_t)nthr * 16)
        __builtin_prefetch(W + i, 0, 0);
    }

    if (!last) {  // zero-fill next activation buffer (provides K padding)
      for (int i = tid; i < tile * Ko; i += nthr) nxt[i] = 0.f;
    }
    __syncthreads();

    const int nMt = tile >> 4;
    const int nNt = (Cout + 15) >> 4;
    for (int t = wave; t < nMt * nNt; t += nwaves) {
      const int mt = t / nNt, nt = t % nNt;
      const int col = (nt << 4) + l15;
      const bool colok = col < Cout;
      const int colc = colok ? col : (Cout - 1);  // clamped, always in-bounds
      const int arow = (mt << 4) + l15;

      const float bb = Bs[colc];
      v8f acc;
#pragma unroll
      for (int v = 0; v < 8; ++v) acc[v] = bb;  // bias: same col across frag

      const float* ap = cur + arow * Kp + khalf;     // 8B-aligned (Kp%4==0)
      const float* wp = W + (size_t)khalf * Cout + colc;
      const size_t wstep = (size_t)Cout * 4;
      const int kLast = Kp - 4;  // Kp >= 4 always

      // Main loop: fully in-bounds (k+khalf+1 <= Kp-5 < Cin), branchless.
#pragma unroll 2
      for (int k = 0; k < kLast; k += 4) {
        const v2f a = *(const v2f*)ap;
        v2f bf;
        bf[0] = wp[0];
        bf[1] = wp[Cout];
        ap += 4;
        wp += wstep;
        acc = __builtin_amdgcn_wmma_f32_16x16x4_f32(false, a, false, bf,
                                                    (short)0, acc, false,
                                                    false);
      }
      // Peeled tail: A is zero-padded there; clamp weight rows in-bounds.
      {
        const int k0 = kLast + khalf;
        const int kk0 = (k0 < Cin) ? k0 : (Cin - 1);
        const int kk1 = (k0 + 1 < Cin) ? (k0 + 1) : (Cin - 1);
        const v2f a = *(const v2f*)ap;
        v2f bf;
        bf[0] = W[(size_t)kk0 * Cout + colc];
        bf[1] = W[(size_t)kk1 * Cout + colc];
        acc = __builtin_amdgcn_wmma_f32_16x16x4_f32(false, a, false, bf,
                                                    (short)0, acc, false,
                                                    false);
      }

#pragma unroll
      for (int v = 0; v < 8; ++v) acc[v] = fmaxf(acc[v], 0.f);  // ReLU

      if (!last) {
        if (colok) {
#pragma unroll
          for (int v = 0; v < 8; ++v) {
            const int r = (mt << 4) + rhalf + v;
            nxt[r * Ko + col] = acc[v];
          }
        }
      } else {
        if (colok) {
#pragma unroll
          for (int v = 0; v < 8; ++v) {
            const int gr = row0 + (mt << 4) + rhalf + v;
            if (gr < R) {
              const int orow = gr / pool;
              atomicMax((int*)out + (size_t)orow * Cout + col,
                        __float_as_int(acc[v]));
            }
          }
        }
      }
    }
    __syncthreads();
    float* tmp = cur; cur = nxt; nxt = tmp;
  }
}

// ------------------------------- host side ----------------------------------

static void launch_fused(const float* in, int R, const MlpDesc& md, float* out,
                         int pool, int tile, hipStream_t s) {
  int cmax = (md.cin[0] + 3) & ~3;
  for (int l = 0; l < md.nlayers; ++l) {
    const int k = (md.cout[l] + 3) & ~3;
    if (k > cmax) cmax = k;
  }
  const size_t shm = (size_t)2 * tile * cmax * sizeof(float);
  const int grid = (R + tile - 1) / tile;
  fused_mlp_pool<<<grid, 256, shm, s>>>(in, R, tile, md, out, pool, cmax);
}

static inline size_t alignup256(size_t x) { return (x + 255) & ~(size_t)255; }

extern "C" void kernel_launch(void* const* d_in, const int* in_sizes, int n_in,
                              void* d_out, int out_size, void* d_ws,
                              size_t ws_size, hipStream_t stream) {
  (void)in_sizes; (void)n_in; (void)out_size; (void)ws_size;
  const int B = 8, N = 8192;

  // ---- inputs (pytree-flattened setup_inputs order) ----
  int ii = 0;
  const float* xyz = (const float*)d_in[ii++];
  const float *sa1w[3], *sa1b[3], *sa2w[3], *sa2b[3], *sa3w[3], *sa3b[3];
  const float *sa4w[4], *sa4b[4], *sa5w[3], *sa5b[3], *fcw[3], *fcb[3];
  for (int k = 0; k < 3; ++k) sa1w[k] = (const float*)d_in[ii++];
  for (int k = 0; k < 3; ++k) sa1b[k] = (const float*)d_in[ii++];
  for (int k = 0; k < 3; ++k) sa2w[k] = (const float*)d_in[ii++];
  for (int k = 0; k < 3; ++k) sa2b[k] = (const float*)d_in[ii++];
  for (int k = 0; k < 3; ++k) sa3w[k] = (const float*)d_in[ii++];
  for (int k = 0; k < 3; ++k) sa3b[k] = (const float*)d_in[ii++];
  for (int k = 0; k < 4; ++k) sa4w[k] = (const float*)d_in[ii++];
  for (int k = 0; k < 4; ++k) sa4b[k] = (const float*)d_in[ii++];
  for (int k = 0; k < 3; ++k) sa5w[k] = (const float*)d_in[ii++];
  for (int k = 0; k < 3; ++k) sa5b[k] = (const float*)d_in[ii++];
  for (int k = 0; k < 3; ++k) fcw[k] = (const float*)d_in[ii++];
  for (int k = 0; k < 3; ++k) fcb[k] = (const float*)d_in[ii++];

  // ---- workspace layout ----
  char* w = (char*)d_ws;
  size_t off = 0;
  auto alloc = [&](size_t bytes) -> void* {
    void* p = w + off;
    off = alignup256(off + bytes);
    return p;
  };
  int* fidx1 = (int*)alloc((size_t)B * 512 * 4);
  int* fidx2 = (int*)alloc((size_t)B * 64 * 4);
  int* gidx1 = (int*)alloc((size_t)B * 512 * 32 * 4);
  int* gidx2 = (int*)alloc((size_t)B * 64 * 64 * 4);
  int* gidx3 = (int*)alloc((size_t)B * 512 * 64 * 4);
  float* nx1 = (float*)alloc((size_t)B * 512 * 3 * 4);
  float* nx2 = (float*)alloc((size_t)B * 64 * 3 * 4);
  float* feats1 = (float*)alloc((size_t)B * 512 * 32 * 3 * 4);
  float* feats2 = (float*)alloc((size_t)B * 64 * 64 * 131 * 4);
  float* feats3 = (float*)alloc((size_t)B * 512 * 64 * 3 * 4);
  float* feats5 = (float*)alloc((size_t)B * 576 * 259 * 4);
  float* feat = (float*)alloc((size_t)B * 1536 * 4);
  // pooled outputs (contiguous, zeroed every call)
  const int n_p1a = B * 512 * 128, n_p1b = B * 64 * 256, n_p2 = B * 512 * 256;
  const int n_p3 = B * 512, n_p4 = B * 1024, n_fco = B * 256;
  const int n_pooled = n_p1a + n_p1b + n_p2 + n_p3 + n_p4 + n_fco;
  float* pooled = (float*)alloc((size_t)n_pooled * 4);
  float* p1a = pooled;
  float* p1b = p1a + n_p1a;
  float* p2 = p1b + n_p1b;
  float* p3 = p2 + n_p2;
  float* p4 = p3 + n_p3;
  float* fco = p4 + n_p4;

  zero_kernel<<<(n_pooled + 255) / 256, 256, 0, stream>>>(pooled, n_pooled);

  // ---- shared FPS for SA1/SA3: fps(xyz, 512) ----
  fps_kernel<<<B, 256, (size_t)N * 4, stream>>>(xyz, N, 512, fidx1);
  gather_xyz_kernel<<<(B * 512 + 255) / 256, 256, 0, stream>>>(
      xyz, fidx1, N, 512, B * 512, nx1);

  // ---- SA1: radius 0.2, ns 32, MLP [3->64->64->128], pool 32 ----
  ball_query_kernel<<<(B * 512 + 255) / 256, 256, 0, stream>>>(
      xyz, nx1, N, 512, B * 512, 32, 0.04f, gidx1);
  {
    const long long rows = (long long)B * 512 * 32;
    group_feats_kernel<<<(int)((rows + 255) / 256), 256, 0, stream>>>(
        xyz, nullptr, nx1, gidx1, B, N, 512, 32, 0, feats1);
    MlpDesc m{};
    m.nlayers = 3;
    m.cin[0] = 3;  m.cout[0] = 64;  m.w[0] = sa1w[0]; m.b[0] = sa1b[0];
    m.cin[1] = 64; m.cout[1] = 64;  m.w[1] = sa1w[1]; m.b[1] = sa1b[1];
    m.cin[2] = 64; m.cout[2] = 128; m.w[2] = sa1w[2]; m.b[2] = sa1b[2];
    launch_fused(feats1, (int)rows, m, p1a, 32, 32, stream);
  }

  // ---- SA2 on (nx1, p1a): fps 64, radius 0.4, ns 64, MLP [131->128->128->256] ----
  fps_kernel<<<B, 256, (size_t)512 * 4, stream>>>(nx1, 512, 64, fidx2);
  gather_xyz_kernel<<<(B * 64 + 255) / 256, 256, 0, stream>>>(
      nx1, fidx2, 512, 64, B * 64, nx2);
  ball_query_kernel<<<(B * 64 + 255) / 256, 256, 0, stream>>>(
      nx1, nx2, 512, 64, B * 64, 64, 0.16f, gidx2);
  {
    const long long rows = (long long)B * 64 * 64;
    group_feats_kernel<<<(int)((rows + 255) / 256), 256, 0, stream>>>(
        nx1, p1a, nx2, gidx2, B, 512, 64, 64, 128, feats2);
    MlpDesc m{};
    m.nlayers = 3;
    m.cin[0] = 131; m.cout[0] = 128; m.w[0] = sa2w[0]; m.b[0] = sa2b[0];
    m.cin[1] = 128; m.cout[1] = 128; m.w[1] = sa2w[1]; m.b[1] = sa2b[1];
    m.cin[2] = 128; m.cout[2] = 256; m.w[2] = sa2w[2]; m.b[2] = sa2b[2];
    launch_fused(feats2, (int)rows, m, p1b, 64, 32, stream);
  }

  // ---- SA3 on xyz (same fidx1/nx1): radius 0.4, ns 64, MLP [3->128->128->256] ----
  ball_query_kernel<<<(B * 512 + 255) / 256, 256, 0, stream>>>(
      xyz, nx1, N, 512, B * 512, 64, 0.16f, gidx3);
  {
    const long long rows = (long long)B * 512 * 64;
    group_feats_kernel<<<(int)((rows + 255) / 256), 256, 0, stream>>>(
        xyz, nullptr, nx1, gidx3, B, N, 512, 64, 0, feats3);
    MlpDesc m{};
    m.nlayers = 3;
    m.cin[0] = 3;   m.cout[0] = 128; m.w[0] = sa3w[0]; m.b[0] = sa3b[0];
    m.cin[1] = 128; m.cout[1] = 128; m.w[1] = sa3w[1]; m.b[1] = sa3b[1];
    m.cin[2] = 128; m.cout[2] = 256; m.w[2] = sa3w[2]; m.b[2] = sa3b[2];
    launch_fused(feats3, (int)rows, m, p2, 64, 32, stream);
  }

  // ---- SA4: _sa_all(xyz): MLP [3->64->128->256->512], pool over all 8192 ----
  {
    MlpDesc m{};
    m.nlayers = 4;
    m.cin[0] = 3;   m.cout[0] = 64;  m.w[0] = sa4w[0]; m.b[0] = sa4b[0];
    m.cin[1] = 64;  m.cout[1] = 128; m.w[1] = sa4w[1]; m.b[1] = sa4b[1];
    m.cin[2] = 128; m.cout[2] = 256; m.w[2] = sa4w[2]; m.b[2] = sa4b[2];
    m.cin[3] = 256; m.cout[3] = 512; m.w[3] = sa4w[3]; m.b[3] = sa4b[3];
    launch_fused(xyz, B * N, m, p3, N, 32, stream);
  }

  // ---- SA5: _sa_all(cx, cp): MLP [259->256->512->1024], pool over 576 ----
  build_feats5_kernel<<<(B * 576 + 255) / 256, 256, 0, stream>>>(nx2, p1b, nx1,
                                                                 p2, feats5);
  {
    MlpDesc m{};
    m.nlayers = 3;
    m.cin[0] = 259; m.cout[0] = 256;  m.w[0] = sa5w[0]; m.b[0] = sa5b[0];
    m.cin[1] = 256; m.cout[1] = 512;  m.w[1] = sa5w[1]; m.b[1] = sa5b[1];
    m.cin[2] = 512; m.cout[2] = 1024; m.w[2] = sa5w[2]; m.b[2] = sa5b[2];
    launch_fused(feats5, B * 576, m, p4, 576, 16, stream);
  }

  // ---- FC head: [1536->1024->512->256], no pooling (pool=1) ----
  build_feat_kernel<<<(B * 1536 + 255) / 256, 256, 0, stream>>>(p3, p4, feat);
  {
    MlpDesc m{};
    m.nlayers = 3;
    m.cin[0] = 1536; m.cout[0] = 1024; m.w[0] = fcw[0]; m.b[0] = fcb[0];
    m.cin[1] = 1024; m.cout[1] = 512;  m.w[1] = fcw[1]; m.b[1] = fcb[1];
    m.cin[2] = 512;  m.cout[2] = 256;  m.w[2] = fcw[2]; m.b[2] = fcb[2];
    launch_fused(feat, B, m, fco, 1, 16, stream);
  }

  // ---- outputs: x4 = zeros(8,1,3), net = (8,256) ----
  finalize_kernel<<<(2048 + 255) / 256, 256, 0, stream>>>(fco, (float*)d_out);
}